// Model_85615878078986
// MI455X (gfx1250) — compile-verified
//
#include <hip/hip_runtime.h>
#include <hip/hip_bf16.h>
#include <math.h>

// Problem constants from the reference
#define BB 4
#define TT 8
#define NN 4096
#define FF 32
#define CC 32
#define HH 64
#define BT (BB*TT)          // 32
#define JJ (BT*FF)          // 1024 packed columns (bt*32 + f)
#define H3 (3*HH)           // 192
#define EPS 1e-5f
#define ELEMS (BB*TT*NN*FF) // 4,194,304
#define EPC (BB*TT*NN)      // 131,072 elements per BN channel

#define KC 64               // K-chunk staged in LDS per step
#define NCHUNK (NN / KC)    // 64
#define PRS 288             // padded pair-row stride in dwords (256 data + 32 pad)

typedef float v2f __attribute__((ext_vector_type(2)));
typedef float v8f __attribute__((ext_vector_type(8)));

// ---------------------------------------------------------------------------
// 1) BatchNorm statistics: per-block partial sum/sumsq per channel (32 ch).
//    Deterministic tree reduction (no float atomics -> stable across replays).
// ---------------------------------------------------------------------------
__global__ __launch_bounds__(256) void bn_stats(const float* __restrict__ x,
                                                float* __restrict__ part) {
    __shared__ float ls[256], ls2[256];
    const int t = threadIdx.x;
    float s = 0.f, s2 = 0.f;
    const size_t base = (size_t)blockIdx.x * 16384;
    #pragma unroll 4
    for (int i = 0; i < 64; ++i) {
        float v = x[base + (size_t)i * 256 + t];   // channel == t&31 (F=32 contiguous)
        s += v; s2 += v * v;
    }
    ls[t] = s; ls2[t] = s2;
    __syncthreads();
    for (int off = 128; off >= 32; off >>= 1) {    // reduce over the 8 rows per channel
        if (t < off) { ls[t] += ls[t + off]; ls2[t] += ls2[t + off]; }
        __syncthreads();
    }
    if (t < 32) {
        part[blockIdx.x * 64 + t]      = ls[t];
        part[blockIdx.x * 64 + 32 + t] = ls2[t];
    }
}

__global__ void bn_finalize(const float* __restrict__ part,
                            const float* __restrict__ gamma,
                            const float* __restrict__ beta,
                            float* __restrict__ bn) {
    const int f = threadIdx.x;                     // 32 threads
    float s = 0.f, s2 = 0.f;
    for (int b = 0; b < 256; ++b) {
        s  += part[b * 64 + f];
        s2 += part[b * 64 + 32 + f];
    }
    const float inv  = 1.0f / (float)EPC;
    const float mean = s * inv;
    const float var  = s2 * inv - mean * mean;     // biased var, as in reference
    const float sc   = gamma[f] * rsqrtf(var + EPS);
    bn[f]      = sc;
    bn[32 + f] = beta[f] - mean * sc;
}

// ---------------------------------------------------------------------------
// 2) Fold the two linear layers: Wc = W_gcn @ W_ih  [32x192],
//    bc = b_gcn @ W_ih + b_ih  [192]
// ---------------------------------------------------------------------------
__global__ void combine_w(const float* __restrict__ Wg, const float* __restrict__ bg,
                          const float* __restrict__ Wih, const float* __restrict__ bih,
                          float* __restrict__ Wc, float* __restrict__ bc) {
    const int h = threadIdx.x;                     // 192 threads
    for (int f = 0; f < FF; ++f) {
        float s = 0.f;
        #pragma unroll
        for (int c = 0; c < CC; ++c) s += Wg[f * CC + c] * Wih[c * H3 + h];
        Wc[f * H3 + h] = s;
    }
    float s = 0.f;
    #pragma unroll
    for (int c = 0; c < CC; ++c) s += bg[c] * Wih[c * H3 + h];
    bc[h] = s + bih[h];
}

// ---------------------------------------------------------------------------
// 3) Normalize + pack into the WMMA-fragment-native, pair-interleaved layout:
//    Xp[((n>>1)*JJ + j)*2 + (n&1)] = scale[f]*x[b,t,n,f] + shift[f]
//    (K rows 2p,2p+1 zipped element-wise -> a B fragment is one ds_load_b64)
// ---------------------------------------------------------------------------
__global__ __launch_bounds__(256) void bn_pack(const float* __restrict__ x,
                                               const float* __restrict__ bn,
                                               float* __restrict__ X) {
    const size_t g = (size_t)blockIdx.x * 256 + threadIdx.x;   // 0..ELEMS-1
    const int f    = (int)(g & 31);
    const size_t r = g >> 5;                                   // bt*N + n
    const int n    = (int)(r & (NN - 1));
    const int bt   = (int)(r >> 12);
    const int j    = bt * FF + f;
    X[((size_t)(n >> 1) * JJ + j) * 2 + (n & 1)] = x[g] * bn[f] + bn[32 + f];
}

// ---------------------------------------------------------------------------
// 4) GEMM on the WMMA path, full fp32, async-to-LDS B staging:
//    Y[4096 x 1024] = adj[4096 x 4096] @ Xp[4096 x 1024]
//
//    Block = 4 waves sharing one 128-column B panel, staged in LDS in 64-row
//    K-chunks (32 pair-rows), double-buffered. Each wave copies 8 pair-rows
//    (2 x global_load_async_to_lds_b128 each = 16 async ops) per chunk and
//    waits with s_wait_asynccnt 0x10 so the next chunk stays in flight.
//    Pair-rows are padded to 288 dwords so the compute-side ds_load_b64 is
//    conflict-free: lanes 0-15 hit banks 0-31, lanes 16-31 (next pair-row,
//    +288 = +32 mod 64) hit banks 32-63. The pair-interleaved layout means a
//    B fragment is one aligned b64 load feeding v_wmma directly (no repack
//    v_dual_mov traffic).
//
//    Fragment maps (ISA 7.12.2, 32-bit):
//      A 16x4 : lane L -> m=L&15, half=L>>4; vgpr v holds A[m][2*half+v]
//      B 4x16 : lane L -> n=L&15, half=L>>4; vgpr v holds B[2*half+v][n]
//      D 16x16: lane L -> n=L&15; vgpr r holds D[r + 8*(L>>4)][n]
// ---------------------------------------------------------------------------
__global__ __launch_bounds__(128) void gcn_wmma_gemm(const float* __restrict__ A,
                                                     const float* __restrict__ X,
                                                     float* __restrict__ Y) {
    __shared__ float sB[2][(KC / 2) * PRS];        // 2 x 36 KB double-buffered panel

    const int t0   = threadIdx.x;
    const int w    = t0 >> 5;                      // wave 0..3
    const int lane = t0 & 31;
    const int half = lane >> 4;                    // 0/1
    const int l15  = lane & 15;
    const int m0   = (blockIdx.x * 4 + w) * 16;    // 64 M-groups x 4 waves
    const int j0   = blockIdx.y * 128;             // 8 column-groups

    // ---- async copy of one chunk (this wave's 8 pair-rows, 1 KB each) ----
    auto issue_chunk = [&](int c) {
        const int pbase = c * (KC / 2);            // global pair-row base
        #pragma unroll
        for (int rr = 0; rr < 8; ++rr) {
            const int pp = w * 8 + rr;             // local pair-row 0..31
            const float* gsrc = X + ((size_t)(pbase + pp) * JJ + j0) * 2;
            #pragma unroll
            for (int hseg = 0; hseg < 2; ++hseg) {
                unsigned lds = (unsigned)(size_t)
                    &sB[c & 1][pp * PRS + hseg * 128 + 4 * lane];
                unsigned long long g =
                    (unsigned long long)(gsrc + hseg * 128 + 4 * lane);
                asm volatile("global_load_async_to_lds_b128 %0, %1, off"
                             :: "v"(lds), "v"(g) : "memory");
            }
        }
    };

    v8f acc[8] = {};
    const float* arow = A + (size_t)(m0 + l15) * NN + 2 * half; // 8B aligned

    issue_chunk(0);                                // prologue

    for (int c = 0; c < NCHUNK; ++c) {
        // keep next chunk in flight while waiting for current one
        if (c + 1 < NCHUNK) {
            issue_chunk(c + 1);
            asm volatile("s_wait_asynccnt 0x10" ::: "memory");  // <=16: chunk c done
        } else {
            asm volatile("s_wait_asynccnt 0x0" ::: "memory");
        }
        __syncthreads();                           // chunk c visible to all 4 waves

        const float* bp = &sB[c & 1][0];
        const int    kg = c * KC;
        if (kg + 1024 < NN)                        // uniform guard, EXEC unchanged
            __builtin_prefetch(arow + kg + 1024, 0, 1);

        for (int k = 0; k < KC; k += 4) {
            v2f a = *(const v2f*)(arow + kg + k);  // A[m][k+2h], A[m][k+2h+1]
            // pair-row (k/2 + half) holds rows {kg+k+2h, kg+k+2h+1}
            const float* bk = bp + (k / 2 + half) * PRS + 2 * l15;
            #pragma unroll
            for (int jt = 0; jt < 8; ++jt) {
                v2f b = *(const v2f*)(bk + jt * 32);   // one ds_load_b64 fragment
                acc[jt] = __builtin_amdgcn_wmma_f32_16x16x4_f32(
                    false, a, false, b, (short)0, acc[jt], false, false);
            }
        }
        __syncthreads();                           // done reading buf before overwrite
    }

    #pragma unroll
    for (int jt = 0; jt < 8; ++jt)
        #pragma unroll
        for (int r = 0; r < 8; ++r)
            Y[(size_t)(m0 + r + 8 * half) * JJ + j0 + jt * 16 + l15] = acc[jt][r];
}

// ---------------------------------------------------------------------------
// 5) GRU over T=8 + fused dense output. One thread per node; h[64] in VGPRs,
//    weights in LDS (24KB + 48KB), h_new staged through LDS (no scratch).
//    gi is reconstructed on the fly from ya via the fused Wc/bc.
// ---------------------------------------------------------------------------
__global__ __launch_bounds__(128) void gru_out(const float* __restrict__ ya,  // [N][JJ]
                                               const float* __restrict__ Wc,  // [32][192]
                                               const float* __restrict__ bc,  // [192]
                                               const float* __restrict__ Whh, // [64][192]
                                               const float* __restrict__ bhh, // [192]
                                               const float* __restrict__ Wd,  // [64]
                                               const float* __restrict__ bd,  // [1]
                                               float* __restrict__ out) {     // [B][T][N]
    __shared__ float sWc[FF * H3];     // 24 KB
    __shared__ float sWhh[HH * H3];    // 48 KB
    __shared__ float sbc[H3], sbhh[H3], sWd[HH];
    __shared__ float sHn[HH * 128];    // 32 KB: per-thread h_new staging

    const int t0 = threadIdx.x;
    for (int i = t0; i < FF * H3; i += 128) sWc[i]  = Wc[i];
    for (int i = t0; i < HH * H3; i += 128) sWhh[i] = Whh[i];
    for (int i = t0; i < H3;      i += 128) { sbc[i] = bc[i]; sbhh[i] = bhh[i]; }
    if (t0 < HH) sWd[t0] = Wd[t0];
    __syncthreads();

    const int b = blockIdx.y;
    const int n = blockIdx.x * 128 + t0;
    const float bdv = bd[0];

    float h[HH];
    #pragma unroll
    for (int i = 0; i < HH; ++i) h[i] = 0.f;

    const float* yrow = ya + (size_t)n * JJ + b * TT * FF;  // 16B aligned

    for (int t = 0; t < TT; ++t) {
        float yr[FF];
        const float4* yv = (const float4*)(yrow + t * FF);
        #pragma unroll
        for (int q = 0; q < 8; ++q) {
            float4 v = yv[q];
            yr[4*q] = v.x; yr[4*q+1] = v.y; yr[4*q+2] = v.z; yr[4*q+3] = v.w;
        }
        for (int hh = 0; hh < HH; ++hh) {
            float gr  = sbc[hh]        + sbhh[hh];
            float gz  = sbc[HH + hh]   + sbhh[HH + hh];
            float gni = sbc[2*HH + hh];
            float gnh = sbhh[2*HH + hh];
            #pragma unroll
            for (int c = 0; c < FF; ++c) {
                const float v = yr[c];
                gr  += v * sWc[c * H3 + hh];
                gz  += v * sWc[c * H3 + HH + hh];
                gni += v * sWc[c * H3 + 2*HH + hh];
            }
            #pragma unroll
            for (int c = 0; c < HH; ++c) {
                const float v = h[c];
                gr  += v * sWhh[c * H3 + hh];
                gz  += v * sWhh[c * H3 + HH + hh];
                gnh += v * sWhh[c * H3 + 2*HH + hh];
            }
            const float r  = 1.f / (1.f + __expf(-gr));
            const float z  = 1.f / (1.f + __expf(-gz));
            const float nn = tanhf(gni + r * gnh);
            sHn[hh * 128 + t0] = (1.f - z) * nn + z * h[hh];
        }
        float o = bdv;
        #pragma unroll
        for (int i = 0; i < HH; ++i) {
            h[i] = sHn[i * 128 + t0];
            o += h[i] * sWd[i];
        }
        out[(size_t)(b * TT + t) * NN + n] = o;
    }
}

// ---------------------------------------------------------------------------
// Launch
// ---------------------------------------------------------------------------
extern "C" void kernel_launch(void* const* d_in, const int* in_sizes, int n_in,
                              void* d_out, int out_size, void* d_ws, size_t ws_size,
                              hipStream_t stream) {
    const float* x     = (const float*)d_in[0];
    const float* adj   = (const float*)d_in[1];
    const float* gamma = (const float*)d_in[2];
    const float* beta  = (const float*)d_in[3];
    const float* Wg    = (const float*)d_in[4];
    const float* bg    = (const float*)d_in[5];
    const float* Wih   = (const float*)d_in[6];
    const float* Whh   = (const float*)d_in[7];
    const float* bih   = (const float*)d_in[8];
    const float* bhh   = (const float*)d_in[9];
    const float* Wd    = (const float*)d_in[10];
    const float* bd    = (const float*)d_in[11];
    float* out = (float*)d_out;

    float* ws   = (float*)d_ws;
    float* part = ws;                       // 16384
    float* bn   = part + 16384;             // 64
    float* Wc   = bn + 64;                  // 6144
    float* bc   = Wc + 6144;                // 192
    float* X    = bc + 192;                 // 4,194,304 (16 MB), pair-interleaved
    float* Y    = X + (size_t)NN * JJ;      // 4,194,304 (16 MB)

    bn_stats   <<<256, 256, 0, stream>>>(x, part);
    bn_finalize<<<1, 32, 0, stream>>>(part, gamma, beta, bn);
    combine_w  <<<1, H3, 0, stream>>>(Wg, bg, Wih, bih, Wc, bc);
    bn_pack    <<<ELEMS / 256, 256, 0, stream>>>(x, bn, X);
    gcn_wmma_gemm<<<dim3(NN / 16 / 4, JJ / 128), 128, 0, stream>>>(adj, X, Y);
    gru_out    <<<dim3(NN / 128, BB), 128, 0, stream>>>(Y, Wc, bc, Whh, bhh, Wd, bd, out);
}